// PatchEmbedding_17291538334269
// MI455X (gfx1250) — compile-verified
//
#include <hip/hip_runtime.h>

// Submanifold 3x3x3 sparse conv (Cin=6 -> Cout=32) + BatchNorm + ReLU for
// MI455X (gfx1250, wave32). Conv GEMM runs on V_WMMA_F32_16X16X4_F32.

#define GRD 200
#define CIN 6
#define COUT 32
#define EPSV 1e-5f
#define HBITS 21
#define HSIZE (1u << HBITS)
#define HMASK (HSIZE - 1u)

typedef __attribute__((ext_vector_type(2))) float v2f;
typedef __attribute__((ext_vector_type(4))) float v4f;
typedef __attribute__((ext_vector_type(8))) float v8f;

#if defined(__has_builtin)
#if __has_builtin(__builtin_amdgcn_global_load_async_to_lds_b32) && \
    __has_builtin(__builtin_amdgcn_s_wait_asynccnt)
#define USE_ASYNC_LDS 1
#endif
#endif

typedef __attribute__((address_space(1))) int* gint_p;
typedef __attribute__((address_space(3))) int* lint_p;

// ---------------------------------------------------------------------------
// Hash table build: voxel key -> row index (open addressing, linear probe)
// ---------------------------------------------------------------------------
__global__ void init_hash_kernel(int* __restrict__ hkey) {
  unsigned i = blockIdx.x * blockDim.x + threadIdx.x;
  if (i < HSIZE) hkey[i] = -1;
}

__global__ void build_hash_kernel(const int* __restrict__ coords, int N,
                                  int* __restrict__ hkey, int* __restrict__ hval) {
  int i = blockIdx.x * blockDim.x + threadIdx.x;
  if (i >= N) return;
  int x = coords[3 * i], y = coords[3 * i + 1], z = coords[3 * i + 2];
  int key = (x * GRD + y) * GRD + z;
  unsigned h = ((unsigned)key * 2654435761u) >> (32 - HBITS);
  while (true) {
    int old = atomicCAS(&hkey[h], -1, key);
    if (old == -1 || old == key) { hval[h] = i; break; }
    h = (h + 1) & HMASK;
  }
}

// ---------------------------------------------------------------------------
// Fused rulebook-lookup + gather + WMMA conv. One wave per 16-site tile.
//   A (16x4 f32): lanes 0-15 hold K-pair {0,1}, lanes 16-31 hold {2,3}.
//   B (4x16 f32): lane holds column n=lane%16; VGPR0/1 = rows ps*2, ps*2+1.
//   Cin=6 padded to K=8 -> 2 WMMAs per N-tile per offset; 2 N-tiles for Cout=32.
// Lookups resolved first (low half only, broadcast by shuffle), then a fully
// unrolled gather+WMMA stream (108 v_wmma per tile).
// Also emits deterministic per-block (sum, sumsq) partials for BatchNorm.
// ---------------------------------------------------------------------------
__global__ __launch_bounds__(256) void subm_conv_wmma_kernel(
    const float* __restrict__ feat, const int* __restrict__ coords,
    const float* __restrict__ W, const int* __restrict__ hkey,
    const int* __restrict__ hval, float* __restrict__ out,
    float* __restrict__ partials, int N) {
  // lw layout: [k][ntile][kchunk][ps][n][j] -> per-lane B fragment = ds_load_b64
  __shared__ float lw[27 * 256];
#ifdef USE_ASYNC_LDS
  __shared__ float wraw[27 * CIN * COUT];
#endif
  __shared__ float wsum[8][32];
  __shared__ float wss[8][32];

  const int tid = threadIdx.x;

#ifdef USE_ASYNC_LDS
  // Stage raw weights global -> LDS on the async (ASYNCcnt) path.
  for (int e = tid; e < 27 * CIN * COUT; e += 256) {
    __builtin_amdgcn_global_load_async_to_lds_b32(
        (gint_p)(W + e), (lint_p)&wraw[e], 0, 0);
  }
  __builtin_amdgcn_s_wait_asynccnt(0);
  __syncthreads();
#endif

  // Swizzle weights into WMMA B-fragment order.
  for (int e = tid; e < 27 * 256; e += 256) {
    int j  = e & 1;
    int n  = (e >> 1) & 15;
    int ps = (e >> 5) & 1;
    int kc = (e >> 6) & 1;
    int nt = (e >> 7) & 1;
    int k  = e >> 8;
    int r  = kc * 4 + ps * 2 + j;            // input-channel row (0..7, pad >5)
    float v = 0.0f;
    if (r < CIN) {
#ifdef USE_ASYNC_LDS
      v = wraw[(k * CIN + r) * COUT + nt * 16 + n];
#else
      v = W[(k * CIN + r) * COUT + nt * 16 + n];
#endif
    }
    lw[e] = v;
  }
  __syncthreads();

  const int wave = tid >> 5;
  const int lane = tid & 31;
  const int mloc = lane & 15;                // tile row this lane serves
  const int ps   = lane >> 4;                // which K-pair this lane holds
  const int base = (blockIdx.x * 8 + wave) * 16;

  const int p = base + mloc;
  const bool rowvalid = (p < N);
  const int pc = rowvalid ? p : 0;
  const int x = coords[3 * pc], y = coords[3 * pc + 1], z = coords[3 * pc + 2];

  // ---- Phase 1: resolve all 27 neighbor indices (low half probes, shuffle
  //      broadcast to the high half -> no redundant hash traffic).
  int nidx[27];
#pragma unroll
  for (int k = 0; k < 27; ++k) {
    const int oz = k % 3 - 1;
    const int oy = (k / 3) % 3 - 1;
    const int ox = k / 9 - 1;
    int nn = -1;
    if (ps == 0) {
      const int nx = x + ox, ny = y + oy, nz = z + oz;
      if (rowvalid && nx >= 0 && nx < GRD && ny >= 0 && ny < GRD &&
          nz >= 0 && nz < GRD) {
        const int key = (nx * GRD + ny) * GRD + nz;
        unsigned h = ((unsigned)key * 2654435761u) >> (32 - HBITS);
        while (true) {
          const int kk = hkey[h];
          if (kk == key) { nn = hval[h]; break; }
          if (kk == -1) break;
          h = (h + 1) & HMASK;
        }
      }
    }
    nidx[k] = __shfl(nn, mloc);
  }

  // ---- Phase 2: gather + WMMA MAC stream (fully unrolled; EXEC all-ones at
  //      every WMMA since divergence is closed before each call).
  v8f acc0 = {0.f, 0.f, 0.f, 0.f, 0.f, 0.f, 0.f, 0.f};   // cols 0..15
  v8f acc1 = {0.f, 0.f, 0.f, 0.f, 0.f, 0.f, 0.f, 0.f};   // cols 16..31

#pragma unroll
  for (int k = 0; k < 27; ++k) {
    v2f a01 = {0.f, 0.f};                    // K-chunk 0: channels ps*2, ps*2+1
    v2f a45 = {0.f, 0.f};                    // K-chunk 1: channels 4+ps*2 (pad)
    const int nn = nidx[k];
    if (nn >= 0) {
      const float* fp = feat + (long long)nn * CIN;
      a01 = *(const v2f*)(fp + ps * 2);
      if (ps == 0) a45 = *(const v2f*)(fp + 4);
    }

    const float* lb = &lw[k * 256 + ps * 32 + mloc * 2];
    const v2f b00 = *(const v2f*)(lb + 0);   // ntile0, kchunk0
    const v2f b01 = *(const v2f*)(lb + 64);  // ntile0, kchunk1
    const v2f b10 = *(const v2f*)(lb + 128); // ntile1, kchunk0
    const v2f b11 = *(const v2f*)(lb + 192); // ntile1, kchunk1

    acc0 = __builtin_amdgcn_wmma_f32_16x16x4_f32(false, a01, false, b00,
                                                 (short)0, acc0, false, false);
    acc0 = __builtin_amdgcn_wmma_f32_16x16x4_f32(false, a45, false, b01,
                                                 (short)0, acc0, false, false);
    acc1 = __builtin_amdgcn_wmma_f32_16x16x4_f32(false, a01, false, b10,
                                                 (short)0, acc1, false, false);
    acc1 = __builtin_amdgcn_wmma_f32_16x16x4_f32(false, a45, false, b11,
                                                 (short)0, acc1, false, false);
  }

  // D layout: lane holds column mloc; VGPR v -> tile row v + 8*ps.
  float s0 = 0.f, q0 = 0.f, s1 = 0.f, q1 = 0.f;
#pragma unroll
  for (int v = 0; v < 8; ++v) {
    const int row = base + v + 8 * ps;
    const float c0 = acc0[v], c1 = acc1[v];
    if (row < N) {
      out[(long long)row * COUT + mloc] = c0;
      out[(long long)row * COUT + 16 + mloc] = c1;
    }
    s0 += c0; q0 += c0 * c0;
    s1 += c1; q1 += c1 * c1;
  }

  // Lanes L and L+16 hold the same channel -> fold halves (wave32 shuffle).
  s0 += __shfl_down(s0, 16);
  q0 += __shfl_down(q0, 16);
  s1 += __shfl_down(s1, 16);
  q1 += __shfl_down(q1, 16);
  if (ps == 0) {
    wsum[wave][mloc]      = s0;
    wsum[wave][16 + mloc] = s1;
    wss[wave][mloc]       = q0;
    wss[wave][16 + mloc]  = q1;
  }
  __syncthreads();

  // Fixed-order per-block combine -> deterministic partials (no float atomics).
  if (tid < 32) {
    float s = 0.f, q = 0.f;
#pragma unroll
    for (int w = 0; w < 8; ++w) { s += wsum[w][tid]; q += wss[w][tid]; }
    partials[(long long)blockIdx.x * 64 + tid]      = s;
    partials[(long long)blockIdx.x * 64 + 32 + tid] = q;
  }
}

// ---------------------------------------------------------------------------
// Deterministic fixed-order reduction of block partials -> per-channel
// scale = gamma * rsqrt(var+eps), bias = beta - mean * scale.
// ---------------------------------------------------------------------------
__global__ void reduce_stats_kernel(const float* __restrict__ partials, int nblk,
                                    const float* __restrict__ gamma,
                                    const float* __restrict__ beta,
                                    float* __restrict__ stats, int N) {
  __shared__ float tot[64];
  const int t = threadIdx.x;  // 64 threads
  float a = 0.f;
  for (int b = 0; b < nblk; ++b) a += partials[(long long)b * 64 + t];
  tot[t] = a;
  __syncthreads();
  if (t < 32) {
    const float inv_n = 1.0f / (float)N;
    const float mean = tot[t] * inv_n;
    const float var = tot[32 + t] * inv_n - mean * mean;
    const float sc = rsqrtf(var + EPSV) * gamma[t];
    stats[t] = sc;
    stats[32 + t] = beta[t] - mean * sc;
  }
}

// ---------------------------------------------------------------------------
// BatchNorm apply + ReLU (in place on d_out), float4 vectorized (b128 ops).
// ---------------------------------------------------------------------------
__global__ void bn_relu_kernel(float* __restrict__ out,
                               const float* __restrict__ stats, long long nvec) {
  v4f* o4 = (v4f*)out;
  long long i = (long long)blockIdx.x * blockDim.x + threadIdx.x;
  const long long stride = (long long)gridDim.x * blockDim.x;
  for (; i < nvec; i += stride) {
    v4f xv = o4[i];
    const int c = (int)((i * 4) & 31);
#pragma unroll
    for (int j = 0; j < 4; ++j) {
      xv[j] = fmaxf(fmaf(xv[j], stats[c + j], stats[32 + c + j]), 0.0f);
    }
    o4[i] = xv;
  }
}

// ---------------------------------------------------------------------------
extern "C" void kernel_launch(void* const* d_in, const int* in_sizes, int n_in,
                              void* d_out, int out_size, void* d_ws, size_t ws_size,
                              hipStream_t stream) {
  const float* feat  = (const float*)d_in[0];
  const int*   coords = (const int*)d_in[1];
  const float* W     = (const float*)d_in[2];
  const float* gamma = (const float*)d_in[3];
  const float* beta  = (const float*)d_in[4];
  float* out = (float*)d_out;

  const int N = in_sizes[0] / CIN;
  const int nblk = (N + 127) / 128;  // 8 waves * 16 sites per block

  char* ws = (char*)d_ws;
  int* hkey = (int*)ws;                                   // 8 MB
  int* hval = (int*)(ws + (size_t)HSIZE * 4);             // 8 MB
  float* partials = (float*)(ws + (size_t)HSIZE * 8);     // nblk*64 floats
  float* stats = (float*)(ws + (size_t)HSIZE * 8 + (size_t)nblk * 64 * 4);

  init_hash_kernel<<<(HSIZE + 255) / 256, 256, 0, stream>>>(hkey);
  build_hash_kernel<<<(N + 255) / 256, 256, 0, stream>>>(coords, N, hkey, hval);
  subm_conv_wmma_kernel<<<nblk, 256, 0, stream>>>(feat, coords, W, hkey, hval,
                                                  out, partials, N);
  reduce_stats_kernel<<<1, 64, 0, stream>>>(partials, nblk, gamma, beta, stats, N);

  const long long nvec = (long long)N * COUT / 4;
  bn_relu_kernel<<<2048, 256, 0, stream>>>(out, stats, nvec);
}